// FocalLoss_21569325760612
// MI455X (gfx1250) — compile-verified
//
#include <hip/hip_runtime.h>
#include <hip/hip_bf16.h>

// Focal loss (BCE-with-logits variant) over [B,2] logits -> scalar mean.
// Memory-bound streaming reduction: 192 MB @ 23.3 TB/s ~= 8.2 us floor.
// WMMA not applicable (no matmul structure). gfx1250 paths: b128 nontemporal
// loads, global_prefetch_b8, wave32 ds-swizzle shuffles, split-counter waits.
// This round: replace libdevice log1pf/__powf hot-path bloat with raw
// v_exp_f32/v_log_f32 so the inner loop stays bandwidth-bound, not issue-bound.

typedef float v4f __attribute__((ext_vector_type(4)));
typedef int   v2i __attribute__((ext_vector_type(2)));

#define THREADS 256
#define BLOCKS  2048
#define WARPS   (THREADS / 32)   // wave32: 8 waves per 256-thread block

// log(sigmoid(z)) = min(z,0) - log(1 + exp(-|z|)).
// Using __logf(1+t) instead of log1pf(t): t in (0,1]; where 1+t loses bits of
// t, the log term is negligible vs the min(z,0) term, so bce error is ~ulp.
__device__ __forceinline__ float log_sigmoid_f(float z) {
    float t = __expf(-fabsf(z));               // v_exp_f32
    return fminf(z, 0.0f) - __logf(1.0f + t);  // v_log_f32
}

__device__ __forceinline__ float row_loss(float a, float b, int yt,
                                          float w0, float w1, int gamma) {
    float x  = fmaxf(a, b);
    // jnp.argmax: FIRST max wins ties -> class 0 on tie -> strict b>a
    float wp = (b > a) ? w1 : w0;
    bool pos = (yt != 0);
    // fold: bce = -(yt ? w : 1) * log_sigmoid(yt ? x : -x)
    float z   = pos ?  x : -x;
    float ww  = pos ? wp : 1.0f;
    float bce = -ww * log_sigmoid_f(z);        // bce >= 0 always
    float m   = 1.0f - __expf(-bce);           // in [0,1)
    float mod = (gamma == 2) ? (m * m)         // hot path (uniform branch)
                             : __powf(m, (float)gamma);
    return mod * bce;
}

__global__ __launch_bounds__(THREADS)
void focal_partial_kernel(const v4f* __restrict__ pred4,   // B/2 float4 (2 rows each)
                          const v2i* __restrict__ true2,   // B/2 int2
                          const float* __restrict__ weights,
                          const int* __restrict__ gamma_p,
                          float* __restrict__ partials,    // [BLOCKS]
                          int npairs, int nrows)
{
    const float w0  = weights[0];
    const float w1  = weights[1];
    const int gamma = *gamma_p;

    float acc = 0.0f;
    const int stride = gridDim.x * blockDim.x;
    for (int i = blockIdx.x * blockDim.x + threadIdx.x; i < npairs; i += stride) {
        // run the stream ahead: speculative global_prefetch_b8 (dropped if OOB)
        __builtin_prefetch(&pred4[i + stride], 0, 0);
        // single-pass stream: nontemporal b128/b64 loads, don't thrash L2
        v4f p = __builtin_nontemporal_load(&pred4[i]);
        v2i t = __builtin_nontemporal_load(&true2[i]);
        acc += row_loss(p.x, p.y, t.x, w0, w1, gamma);
        acc += row_loss(p.z, p.w, t.y, w0, w1, gamma);
    }
    // odd-B tail row (cold; not hit for B=16M)
    if ((nrows & 1) && blockIdx.x == 0 && threadIdx.x == 0) {
        const float* pf = (const float*)pred4;
        const int*   tf = (const int*)true2;
        const int last  = nrows - 1;
        acc += row_loss(pf[2 * last], pf[2 * last + 1], tf[last], w0, w1, gamma);
    }

    // wave32 butterfly (all threads reach here -> EXEC full)
    #pragma unroll
    for (int off = 16; off > 0; off >>= 1)
        acc += __shfl_xor(acc, off, 32);

    __shared__ float smem[WARPS];
    const int lane = threadIdx.x & 31;
    const int wave = threadIdx.x >> 5;
    if (lane == 0) smem[wave] = acc;
    __syncthreads();

    if (wave == 0) {                       // whole first wave: EXEC all-1s
        float v = (lane < WARPS) ? smem[lane] : 0.0f;
        #pragma unroll
        for (int off = 16; off > 0; off >>= 1)
            v += __shfl_xor(v, off, 32);
        if (lane == 0) partials[blockIdx.x] = v;   // deterministic per-block partial
    }
}

__global__ __launch_bounds__(THREADS)
void focal_final_kernel(const float* __restrict__ partials, int n,
                        float* __restrict__ out, int nrows)
{
    double acc = 0.0;
    for (int i = threadIdx.x; i < n; i += THREADS)
        acc += (double)partials[i];

    __shared__ double s[THREADS];
    s[threadIdx.x] = acc;
    __syncthreads();
    #pragma unroll
    for (int off = THREADS / 2; off > 0; off >>= 1) {
        if (threadIdx.x < off) s[threadIdx.x] += s[threadIdx.x + off];
        __syncthreads();
    }
    if (threadIdx.x == 0)
        out[0] = (float)(s[0] / (double)nrows);
}

extern "C" void kernel_launch(void* const* d_in, const int* in_sizes, int n_in,
                              void* d_out, int out_size, void* d_ws, size_t ws_size,
                              hipStream_t stream) {
    const v4f*   pred4   = (const v4f*)d_in[0];     // y_pred  f32 [B,2] -> B/2 float4
    const v2i*   true2   = (const v2i*)d_in[1];     // y_true  i32 [B]   -> B/2 int2
    const float* weights = (const float*)d_in[2];   // [2] f32
    const int*   gamma   = (const int*)d_in[3];     // scalar int (1-elem array)

    const int nrows  = in_sizes[1];                 // B
    const int npairs = nrows >> 1;
    float* partials  = (float*)d_ws;                // BLOCKS * 4 B = 8 KB scratch

    focal_partial_kernel<<<BLOCKS, THREADS, 0, stream>>>(
        pred4, true2, weights, gamma, partials, npairs, nrows);
    focal_final_kernel<<<1, THREADS, 0, stream>>>(
        partials, BLOCKS, (float*)d_out, nrows);
}